// end_point_29042568856291
// MI455X (gfx1250) — compile-verified
//
#include <hip/hip_runtime.h>
#include <hip/hip_bf16.h>

// Squared pairwise distance as a rank-4 GEMM on the WMMA unit:
//   sq(i,j) = [xi, yi, ni, 1] . [-2xj, -2yj, 1, nj]   with ni = xi^2+yi^2
// One V_WMMA_F32_16X16X4_F32 per 16x16 output tile. The 512 MB output stream
// (> 192 MB L2) is written with non-temporal stores. Distance and reciprocal
// come from a single v_rsq_f32:  inv = rsq(sq),  d = sq * rsq(sq) = sqrt(sq).

typedef float v2f __attribute__((ext_vector_type(2)));
typedef float v8f __attribute__((ext_vector_type(8)));

#define N_AGENTS 1024
#define SLICE_IN_FLOATS (N_AGENTS * 2)          // [N,2] points per (b,c,t) slice
#define SLICE_OUT_FLOATS (N_AGENTS * N_AGENTS)  // one 1024x1024 matrix

__global__ __launch_bounds__(256) void pairwise_dist_wmma(const float* __restrict__ S,
                                                          float* __restrict__ out) {
    const int lane = threadIdx.x & 31;
    const int wave = threadIdx.x >> 5;   // 8 waves per block
    const int s    = blockIdx.x >> 6;    // slice id: 0..63  ( = (b*2+c)*8 + t )
    const int iblk = blockIdx.x & 63;    // 16-row block within the 1024 rows

    const int b = s >> 4;
    const int c = (s >> 3) & 1;
    const int t = s & 7;

    const float2* __restrict__ P =
        (const float2*)(S + (size_t)s * SLICE_IN_FLOATS);   // 1024 (x,y) points

    // out shape [B,4,T,N,N]; A at channel c, A_inv at channel c+2
    float* __restrict__ outA =
        out + ((size_t)((b * 4 + c) * 8 + t)) * SLICE_OUT_FLOATS;
    float* __restrict__ outI = outA + (size_t)16 * SLICE_OUT_FLOATS; // +2 channels * T

    const int li  = lane & 15;
    const bool lo = lane < 16;

    // ---- A operand: 16x4 f32. lanes 0-15 hold K=0,1 = (xi,yi);
    //      lanes 16-31 hold K=2,3 = (ni, 1).
    const int i0 = iblk * 16;
    float2 pi = P[i0 + li];
    v2f a;
    a.x = lo ? pi.x : (pi.x * pi.x + pi.y * pi.y);
    a.y = lo ? pi.y : 1.0f;

    const int ibase = i0 + (lo ? 0 : 8);   // C-matrix row base for this lane half

    // Each wave covers a 16x128 strip: 8 j-tiles of 16 columns.
    for (int jt = 0; jt < 8; ++jt) {
        const int j0 = wave * 128 + jt * 16;
        float2 pj = P[j0 + li];

        // ---- B operand: 4x16 f32. lanes 0-15 hold K=0,1 rows = (-2xj,-2yj);
        //      lanes 16-31 hold K=2,3 rows = (1, nj).
        v2f bm;
        bm.x = lo ? (-2.0f * pj.x) : 1.0f;
        bm.y = lo ? (-2.0f * pj.y) : (pj.x * pj.x + pj.y * pj.y);

        v8f acc = {0.f, 0.f, 0.f, 0.f, 0.f, 0.f, 0.f, 0.f};
        acc = __builtin_amdgcn_wmma_f32_16x16x4_f32(
            /*neg_a=*/false, a, /*neg_b=*/false, bm,
            /*c_mod=*/(short)0, acc, /*reuse_a=*/false, /*reuse_b=*/false);

        const int jc = j0 + li;
#pragma unroll
        for (int r = 0; r < 8; ++r) {
            const float sq = acc[r];
            const size_t off = (size_t)(ibase + r) * N_AGENTS + jc;
            const bool  nz  = sq > 0.0f;
            const float rs  = __builtin_amdgcn_rsqf(sq);  // v_rsq_f32
            const float d   = nz ? (sq * rs) : 0.0f;      // sqrt(sq) = sq * rsq(sq)
            const float inv = nz ? rs        : 0.0f;
            __builtin_nontemporal_store(d,   outA + off);
            __builtin_nontemporal_store(inv, outI + off);
        }
    }
}

extern "C" void kernel_launch(void* const* d_in, const int* in_sizes, int n_in,
                              void* d_out, int out_size, void* d_ws, size_t ws_size,
                              hipStream_t stream) {
    (void)in_sizes; (void)n_in; (void)d_ws; (void)ws_size; (void)out_size;
    const float* S = (const float*)d_in[0];   // [4,2,8,1024,2] f32
    float* out = (float*)d_out;               // [4,4,8,1024,1024] f32

    // 64 slices * 64 row-blocks = 4096 workgroups, 256 threads (8 waves) each.
    dim3 grid(64 * 64);
    dim3 block(256);
    pairwise_dist_wmma<<<grid, block, 0, stream>>>(S, out);
}